// RoIAlign_17660905521563
// MI455X (gfx1250) — compile-verified
//
#include <hip/hip_runtime.h>

// ---------------------------------------------------------------------------
// RoIAlign forward for MI455X (gfx1250, wave32).
//
// Memory-bound gather kernel: 512 rois x 256 ch x 49 bins, 16 weighted
// gathers per output. Input (164 MB) is L2-resident (192 MB L2), HBM floor
// ~8us @ 23.3 TB/s. We use the CDNA5 async global->LDS gather path
// (ASYNCcnt) to double-buffer the 16 gathers per output across iterations.
// ---------------------------------------------------------------------------

#ifndef __has_builtin
#define __has_builtin(x) 0
#endif

#if defined(__gfx1250__) && \
    __has_builtin(__builtin_amdgcn_global_load_async_to_lds_b32) && \
    __has_builtin(__builtin_amdgcn_s_wait_asynccnt)
#define RA_ASYNC 1
#else
#define RA_ASYNC 0
#endif

namespace {
constexpr int   kOutHW   = 7;
constexpr int   kBins    = kOutHW * kOutHW;   // 49
constexpr int   kSR      = 2;                 // sampling ratio
constexpr int   kC       = 256;
constexpr int   kH       = 200;
constexpr int   kW       = 200;
constexpr int   kHW      = kH * kW;
constexpr float kScale   = 0.25f;
constexpr int   kThreads = 256;               // 8 wave32 waves
constexpr int   kTotal   = kC * kBins;        // 12544 = 49 * 256 exactly
constexpr int   kIters   = kTotal / kThreads; // 49, uniform (no remainder)
}  // namespace

#if RA_ASYNC
typedef __attribute__((address_space(1))) int g_int_t;
typedef __attribute__((address_space(3))) int lds_int_t;

// Generic->AS1: flat global address == AS1 address (64-bit).
__device__ __forceinline__ g_int_t* to_glob(const void* p) {
  return (g_int_t*)(unsigned long long)p;
}
// Generic->AS3: flat LDS address low 32 bits are the LDS offset
// (ISA 10.2 aperture rules), and AS3 pointers are 32-bit.
__device__ __forceinline__ lds_int_t* to_lds(const void* p) {
  return (lds_int_t*)(unsigned)(unsigned long long)p;
}
#endif

__global__ __launch_bounds__(kThreads) void roialign_fwd(
    const float* __restrict__ inp,   // (4, 256, 200, 200)
    const float* __restrict__ rois,  // (K, 5) = {b, x1, y1, x2, y2}
    float* __restrict__ out)         // (K, 256, 7, 7)
{
  // Per-bin table: 16 corners, .x = bit-cast spatial offset, .y = weight/4.
  __shared__ float2 s_tab[kBins * 16];      // 6.3 KB
  __shared__ int    s_base;
#if RA_ASYNC
  __shared__ __align__(16) float s_stage[2][kThreads * 16];  // 32 KB
#endif

  const int k   = blockIdx.x;
  const int tid = threadIdx.x;

  // -------------------- setup: sample offsets & weights --------------------
  if (tid == 0) {
    s_base = (int)rois[k * 5 + 0] * (kC * kHW);
  }
  if (tid < kBins * kSR * kSR) {            // 196 threads: one (bin, sample)
    const int bin = tid >> 2;
    const int smp = tid & 3;
    const int ph  = bin / kOutHW;
    const int pw  = bin - ph * kOutHW;
    const int iy  = smp >> 1;
    const int ix  = smp & 1;

    const float* r = rois + k * 5;
    const float start_w = r[1] * kScale - 0.5f;
    const float start_h = r[2] * kScale - 0.5f;
    const float roi_w   = r[3] * kScale - 0.5f - start_w;
    const float roi_h   = r[4] * kScale - 0.5f - start_h;
    const float bin_h   = roi_h / (float)kOutHW;
    const float bin_w   = roi_w / (float)kOutHW;

    float y = start_h + ((float)ph + ((float)iy + 0.5f) / (float)kSR) * bin_h;
    float x = start_w + ((float)pw + ((float)ix + 0.5f) / (float)kSR) * bin_w;
    y = fmaxf(y, 0.0f);
    x = fmaxf(x, 0.0f);

    int yl = (int)y;
    int yh = (yl >= kH - 1) ? (kH - 1) : (yl + 1);
    yl     = (yl >= kH - 1) ? (kH - 1) : yl;
    const float ly = y - (float)yl, hy = 1.0f - ly;

    int xl = (int)x;
    int xh = (xl >= kW - 1) ? (kW - 1) : (xl + 1);
    xl     = (xl >= kW - 1) ? (kW - 1) : xl;
    const float lx = x - (float)xl, hx = 1.0f - lx;

    const int base = bin * 16 + smp * 4;
    s_tab[base + 0] = make_float2(__int_as_float(yl * kW + xl), 0.25f * hy * hx);
    s_tab[base + 1] = make_float2(__int_as_float(yl * kW + xh), 0.25f * hy * lx);
    s_tab[base + 2] = make_float2(__int_as_float(yh * kW + xl), 0.25f * ly * hx);
    s_tab[base + 3] = make_float2(__int_as_float(yh * kW + xh), 0.25f * ly * lx);
  }
  __syncthreads();

  const float* __restrict__ src = inp + s_base;
  float* __restrict__ dst       = out + k * kTotal;

  // Linear output index i = c*49 + bin; i = it*256 + tid -> coalesced stores.
  int cc = tid / kBins;
  int bb = tid - cc * kBins;

#if RA_ASYNC
  // ------------- async global->LDS gathers, double buffered ---------------
  {
    const float*  p   = src + cc * kHW;
    const float2* tab = &s_tab[bb * 16];
#pragma unroll
    for (int j = 0; j < 16; ++j) {
      __builtin_amdgcn_global_load_async_to_lds_b32(
          to_glob(p + __float_as_int(tab[j].x)),
          to_lds(&s_stage[0][tid * 16 + j]), 0, 0);
    }
  }
#pragma unroll 1
  for (int it = 0; it < kIters; ++it) {
    // next linear index: += 256 -> c += 5, bin += 11 (mod 49 carry)
    int cn = cc + 5, bn = bb + 11;
    if (bn >= kBins) { bn -= kBins; ++cn; }
    const int cur = it & 1;

    if (it + 1 < kIters) {
      const float*  p   = src + cn * kHW;
      const float2* tab = &s_tab[bn * 16];
#pragma unroll
      for (int j = 0; j < 16; ++j) {
        __builtin_amdgcn_global_load_async_to_lds_b32(
            to_glob(p + __float_as_int(tab[j].x)),
            to_lds(&s_stage[cur ^ 1][tid * 16 + j]), 0, 0);
      }
      __builtin_amdgcn_s_wait_asynccnt(16);  // current 16 done, next in flight
    } else {
      __builtin_amdgcn_s_wait_asynccnt(0);
    }

    const float4* v4  = (const float4*)&s_stage[cur][tid * 16];
    const float2* tab = &s_tab[bb * 16];
    float acc = 0.0f;
#pragma unroll
    for (int q = 0; q < 4; ++q) {
      const float4 v = v4[q];
      acc += tab[4 * q + 0].y * v.x;
      acc += tab[4 * q + 1].y * v.y;
      acc += tab[4 * q + 2].y * v.z;
      acc += tab[4 * q + 3].y * v.w;
    }
    dst[cc * kBins + bb] = acc;
    cc = cn; bb = bn;
  }
#else
  // ----------------- fallback: direct gathers + prefetch ------------------
#pragma unroll 1
  for (int it = 0; it < kIters; ++it) {
    const float*  p   = src + cc * kHW;
    const float2* tab = &s_tab[bb * 16];
    float acc = 0.0f;
#pragma unroll
    for (int j = 0; j < 16; ++j) {
      const float2 t = tab[j];
      acc += t.y * p[__float_as_int(t.x)];
    }
    dst[cc * kBins + bb] = acc;

    int cn = cc + 5, bn = bb + 11;
    if (bn >= kBins) { bn -= kBins; ++cn; }
    if (it + 1 < kIters) {
      __builtin_prefetch(src + cn * kHW + __float_as_int(s_tab[bn * 16].x), 0, 0);
    }
    cc = cn; bb = bn;
  }
#endif
}

extern "C" void kernel_launch(void* const* d_in, const int* in_sizes, int n_in,
                              void* d_out, int out_size, void* d_ws, size_t ws_size,
                              hipStream_t stream) {
  const float* inp  = (const float*)d_in[0];   // (4,256,200,200) fp32
  const float* rois = (const float*)d_in[1];   // (K,5) fp32
  float*       outp = (float*)d_out;           // (K,256,7,7) fp32
  const int K = in_sizes[1] / 5;               // 512
  roialign_fwd<<<dim3(K), dim3(kThreads), 0, stream>>>(inp, rois, outp);
}